// _BaseTransducer_70007966925091
// MI455X (gfx1250) — compile-verified
//
#include <hip/hip_runtime.h>

typedef __attribute__((ext_vector_type(2))) float v2f;
typedef __attribute__((ext_vector_type(8))) float v8f;
typedef __attribute__((ext_vector_type(4))) float f4v;

#define B_ 4
#define M_ 256
#define N_ 128
#define D_ 512
#define C_ 1024

// out[r, c] = sum_d A[r, d] * W[c, d] (+ bias[c] if bias != nullptr)
// A: [rows, D] row-major; W: [C, D] row-major; out: [rows, C] row-major.
// One wave (32 lanes) computes one 16x16 tile of out via V_WMMA_F32_16X16X4_F32.
__global__ void __launch_bounds__(32)
joiner_proj_wmma(const float* __restrict__ A,
                 const float* __restrict__ W,
                 const float* __restrict__ bias,
                 float* __restrict__ out) {
    const int c0   = blockIdx.x << 4;   // output column tile
    const int r0   = blockIdx.y << 4;   // output row tile
    const int lane = threadIdx.x;       // 0..31, full wave, EXEC all-1s
    const int lo   = lane & 15;
    const int hi   = lane >> 4;

    // A-matrix 16x4 f32 layout: lanes 0-15 -> K={0,1}, lanes 16-31 -> K={2,3};
    // row M = lane%16. Per-lane b64 load of 2 consecutive k values.
    const float* arow = A + (size_t)(r0 + lo) * D_ + (hi << 1);
    // B-matrix 4x16: B[k,n] = W[c0+n, k]; lane holds column n = lane%16.
    const float* brow = W + (size_t)(c0 + lo) * D_ + (hi << 1);

    // C/D 16x16 f32 layout: vgpr r, lane -> M = r + 8*(lane>>4), N = lane%16.
    // Bias depends only on N -> same value in all 8 accumulator VGPRs.
    float binit = bias ? bias[c0 + lo] : 0.0f;
    v8f acc;
#pragma unroll
    for (int i = 0; i < 8; ++i) acc[i] = binit;

    for (int k = 0; k < D_; k += 16) {
#pragma unroll
        for (int kk = 0; kk < 16; kk += 4) {
            v2f a = *(const v2f*)(arow + k + kk);
            v2f b = *(const v2f*)(brow + k + kk);
            // 8 args: (neg_a, A, neg_b, B, c_mod, C, reuse_a, reuse_b)
            acc = __builtin_amdgcn_wmma_f32_16x16x4_f32(
                false, a, false, b, (short)0, acc, false, false);
        }
    }

    float* orow = out + (size_t)r0 * C_ + c0 + lo;
#pragma unroll
    for (int r = 0; r < 8; ++r) {
        orow[(size_t)(r + (hi << 3)) * C_] = acc[r];
    }
}

// out[b,m,n,c] = E[b*M+m, c] + Dd[b*N+n, c]
// One block per (b, m): thread t keeps its float4 of the E row in a register,
// loops over n streaming Dd rows (L2-resident) and issuing NT b128 stores.
__global__ void __launch_bounds__(256)
joiner_bcast_add(const float* __restrict__ E,   // [B*M, C]
                 const float* __restrict__ Dd,  // [B*N, C]
                 float* __restrict__ out) {     // [B, M, N, C]
    const int m = blockIdx.x;
    const int b = blockIdx.y;
    const int t = threadIdx.x;                  // 0..255 -> C/4 float4 slots

    const f4v* erow = (const f4v*)(E + (size_t)(b * M_ + m) * C_);
    const f4v  ev   = erow[t];

    const f4v* dbase = (const f4v*)(Dd + (size_t)b * N_ * C_);
    f4v* obase = (f4v*)(out + ((size_t)(b * M_ + m) * N_) * C_);

#pragma unroll 4
    for (int n = 0; n < N_; ++n) {
        f4v dv = dbase[(size_t)n * (C_ / 4) + t];
        f4v r;
        r.x = ev.x + dv.x;
        r.y = ev.y + dv.y;
        r.z = ev.z + dv.z;
        r.w = ev.w + dv.w;
        __builtin_nontemporal_store(r, &obase[(size_t)n * (C_ / 4) + t]);
    }
}

extern "C" void kernel_launch(void* const* d_in, const int* in_sizes, int n_in,
                              void* d_out, int out_size, void* d_ws, size_t ws_size,
                              hipStream_t stream) {
    const float* enc  = (const float*)d_in[0];  // [B, M, D]
    const float* dec  = (const float*)d_in[1];  // [B, N, D]
    const float* W    = (const float*)d_in[2];  // [C, D]
    const float* bias = (const float*)d_in[3];  // [C]
    float* out = (float*)d_out;                 // [B, M, N, C]

    float* E  = (float*)d_ws;                   // [B*M, C] = 4 MB
    float* Dd = E + (size_t)B_ * M_ * C_;       // [B*N, C] = 2 MB

    // E = enc @ W^T
    joiner_proj_wmma<<<dim3(C_ / 16, (B_ * M_) / 16), dim3(32), 0, stream>>>(
        enc, W, nullptr, E);
    // Dd = dec @ W^T + bias
    joiner_proj_wmma<<<dim3(C_ / 16, (B_ * N_) / 16), dim3(32), 0, stream>>>(
        dec, W, bias, Dd);
    // out = E broadcast+ Dd  (store-bound: 512 MB NT stores ~= 22 us at 23.3 TB/s)
    joiner_bcast_add<<<dim3(M_, B_), dim3(256), 0, stream>>>(E, Dd, out);
}